// JITScriptKernelizedAttention_4999341932648
// MI455X (gfx1250) — compile-verified
//
#include <hip/hip_runtime.h>
#include <hip/hip_bf16.h>

// Problem constants (fixed by the reference setup_inputs)
#define BATCH 2
#define HEADS 32
#define BH    (BATCH * HEADS)   // 64
#define SQ    2048
#define SKV   8192
#define DD    128
#define SPLIT 8                  // kv chunks per head in phase 1
#define CHUNK (SKV / SPLIT)      // 1024
#define QTILE 128                // q rows per workgroup in phase 2
#define EPSV  1e-6f

typedef __attribute__((ext_vector_type(2))) float v2f;
typedef __attribute__((ext_vector_type(8))) float v8f;

__device__ __forceinline__ float phi(float x) {
    // elu(x) + 1  (alpha = 1)
    return x > 0.0f ? x + 1.0f : __expf(x);
}

// ---------------------------------------------------------------------------
// Kernel 0: zero the KV / Z accumulators in workspace (atomic targets)
// ---------------------------------------------------------------------------
__global__ void zero_ws_kernel(float* __restrict__ ws, int n) {
    int i = blockIdx.x * blockDim.x + threadIdx.x;
    if (i < n) ws[i] = 0.0f;
}

// ---------------------------------------------------------------------------
// Kernel 1: KV[h] += Phi(K_chunk)^T * V_chunk ; Z[h] += sum Phi(K_chunk)
// grid = BH * SPLIT blocks, 256 threads (8 waves).
// Wave w computes d-rows [16w, 16w+16) x all 128 e-cols via f32 WMMA.
// ---------------------------------------------------------------------------
__global__ __launch_bounds__(256) void phase1_kernel(
    const float* __restrict__ K, const float* __restrict__ V,
    float* __restrict__ kv_ws, float* __restrict__ z_ws) {

    const int head  = blockIdx.x / SPLIT;
    const int chunk = blockIdx.x % SPLIT;
    const int wave  = threadIdx.x >> 5;
    const int lane  = threadIdx.x & 31;
    const int r     = lane & 15;   // A: d-row in tile / B: e-col in tile
    const int t     = lane >> 4;   // k-pair selector

    const float* Kh = K + (size_t)head * SKV * DD;
    const float* Vh = V + (size_t)head * SKV * DD;
    const int    k0 = chunk * CHUNK;
    const int    drow = wave * 16 + r;   // this wave's A row (d index)

    v8f   acc[8] = {};
    float zacc = 0.0f;

    for (int kk = 0; kk < CHUNK; kk += 4) {
        const int kb = k0 + kk + 2 * t;
        // A fragment: A[m][j] = Phi(K[kb+j][drow])  (transposed read of K)
        const float a0 = phi(Kh[(size_t)(kb)     * DD + drow]);
        const float a1 = phi(Kh[(size_t)(kb + 1) * DD + drow]);
        v2f a; a.x = a0; a.y = a1;
        zacc += a0 + a1;
#pragma unroll
        for (int et = 0; et < 8; ++et) {
            const int e = et * 16 + r;
            v2f b;
            b.x = Vh[(size_t)(kb)     * DD + e];
            b.y = Vh[(size_t)(kb + 1) * DD + e];
            acc[et] = __builtin_amdgcn_wmma_f32_16x16x4_f32(
                false, a, false, b, (short)0, acc[et], false, false);
        }
    }

    // Merge partial KV tile:  D layout -> row = i + 8*t, col = et*16 + r
    float* kvh = kv_ws + (size_t)head * DD * DD;
#pragma unroll
    for (int et = 0; et < 8; ++et) {
#pragma unroll
        for (int i = 0; i < 8; ++i) {
            const int dd = wave * 16 + i + 8 * t;
            const int ee = et * 16 + r;
            atomicAdd(&kvh[dd * DD + ee], acc[et][i]);
        }
    }
    // Z partial: lanes r and r+16 both accumulated distinct k's for row drow
    atomicAdd(&z_ws[head * DD + drow], zacc);
}

// ---------------------------------------------------------------------------
// Kernel 2: out = (Phi(Q) @ KV) / (Phi(Q) . Z + eps)
// grid = BH * (SQ/QTILE) blocks, 256 threads (8 waves x 16 q-rows each).
// KV staged in LDS (64 KB); B fragments fed from ds_load.
// ---------------------------------------------------------------------------
__global__ __launch_bounds__(256) void phase2_kernel(
    const float* __restrict__ Q, const float* __restrict__ kv_ws,
    const float* __restrict__ z_ws, float* __restrict__ out) {

    const int head = blockIdx.x / (SQ / QTILE);
    const int qblk = blockIdx.x % (SQ / QTILE);
    const int wave = threadIdx.x >> 5;
    const int lane = threadIdx.x & 31;
    const int r    = lane & 15;
    const int t    = lane >> 4;

    __shared__ float kv_s[DD * DD];   // exactly 64 KB

    // Cooperative, coalesced stage of this head's KV into LDS
    const float* kvh = kv_ws + (size_t)head * DD * DD;
    for (int idx = threadIdx.x; idx < DD * DD; idx += 256) {
        kv_s[idx] = kvh[idx];
    }
    __syncthreads();

    const int    q0   = qblk * QTILE + wave * 16;  // wave's first q row
    const int    qrow = q0 + r;
    const float* Qh   = Q + (size_t)head * SQ * DD;
    const float* Zh   = z_ws + head * DD;

    v8f   acc[8] = {};
    float den = 0.0f;

    for (int kb0 = 0; kb0 < DD; kb0 += 4) {
        const int kb = kb0 + 2 * t;
        const float a0 = phi(Qh[(size_t)qrow * DD + kb]);
        const float a1 = phi(Qh[(size_t)qrow * DD + kb + 1]);
        v2f a; a.x = a0; a.y = a1;
        den += a0 * Zh[kb] + a1 * Zh[kb + 1];
#pragma unroll
        for (int et = 0; et < 8; ++et) {
            v2f b;
            b.x = kv_s[(kb)     * DD + et * 16 + r];
            b.y = kv_s[(kb + 1) * DD + et * 16 + r];
            acc[et] = __builtin_amdgcn_wmma_f32_16x16x4_f32(
                false, a, false, b, (short)0, acc[et], false, false);
        }
    }

    // Row q0+r's denominator is split across lanes r and r+16 (k halves)
    den += __shfl_xor(den, 16, 32);

    // Write out: D row = i + 8*t, col = et*16 + r ; divide by denominator
    float* outh = out + (size_t)head * SQ * DD;
#pragma unroll
    for (int i = 0; i < 8; ++i) {
        const int   qq  = q0 + i + 8 * t;
        const float dv  = __shfl(den, i + 8 * t, 32);  // den lives in lane (row&15)
        const float inv = 1.0f / (dv + EPSV);
#pragma unroll
        for (int et = 0; et < 8; ++et) {
            outh[(size_t)qq * DD + et * 16 + r] = acc[et][i] * inv;
        }
    }
}

// ---------------------------------------------------------------------------
extern "C" void kernel_launch(void* const* d_in, const int* in_sizes, int n_in,
                              void* d_out, int out_size, void* d_ws, size_t ws_size,
                              hipStream_t stream) {
    const float* Q = (const float*)d_in[0];   // [2,32,2048,128]
    const float* K = (const float*)d_in[1];   // [2,32,8192,128]
    const float* V = (const float*)d_in[2];   // [2,32,8192,128]
    float* out = (float*)d_out;               // [2,32,2048,128]

    float* kv_ws = (float*)d_ws;                       // BH*128*128 floats
    float* z_ws  = kv_ws + (size_t)BH * DD * DD;       // BH*128 floats
    const int nzero = BH * DD * DD + BH * DD;          // ~1.06M floats (4.2 MB)

    zero_ws_kernel<<<(nzero + 255) / 256, 256, 0, stream>>>(kv_ws, nzero);
    phase1_kernel<<<BH * SPLIT, 256, 0, stream>>>(K, V, kv_ws, z_ws);
    phase2_kernel<<<BH * (SQ / QTILE), 256, 0, stream>>>(Q, kv_ws, z_ws, out);
}